// GNNModel_67327907332268
// MI455X (gfx1250) — compile-verified
//
#include <hip/hip_runtime.h>

// GraphConv x3 for MI455X (gfx1250, wave32).
// Pipeline per layer: [edge gather + atomic scatter-add in L2] -> [fp32 WMMA GEMM].
// out_norm fused into edge coefficient (layers 0/1) or GEMM row-scale (layer 2);
// in_norm fused into GEMM row-scale (layers 0/1) or final epilogue (layer 2).

#define NN 100000      // nodes
#define NE 1600000     // edges
#define KD 128         // inner GEMM dim (all layers have fin-side width 128)

typedef __attribute__((ext_vector_type(2))) float v2f;
typedef __attribute__((ext_vector_type(8))) float v8f;

__global__ void fill_zero_kernel(float* __restrict__ p, int n) {
  int i = blockIdx.x * blockDim.x + threadIdx.x;
  int stride = gridDim.x * blockDim.x;
  for (; i < n; i += stride) p[i] = 0.0f;
}

__global__ void degree_kernel(const int* __restrict__ src, const int* __restrict__ dst,
                              float* __restrict__ out_deg, float* __restrict__ in_deg) {
  int e = blockIdx.x * blockDim.x + threadIdx.x;
  if (e >= NE) return;
  unsafeAtomicAdd(out_deg + src[e], 1.0f);
  unsafeAtomicAdd(in_deg + dst[e], 1.0f);
}

// deg -> clip(deg,1)^-0.5, in place (out_deg and in_deg are contiguous: n = 2*NN)
__global__ void norm_kernel(float* __restrict__ deg, int n) {
  int i = blockIdx.x * blockDim.x + threadIdx.x;
  if (i >= n) return;
  float d = deg[i];
  d = d < 1.0f ? 1.0f : d;
  deg[i] = rsqrtf(d);
}

// agg[dst[e]] += h[src[e]] * ew[e] * (SCALE ? src_scale[src[e]] : 1)
// One thread per (edge, float4 chunk); chunks per edge = 1 << LOG2C; width = 4 << LOG2C.
template <int LOG2C, bool SCALE>
__global__ void edge_aggregate_kernel(const float* __restrict__ h,
                                      const int* __restrict__ src,
                                      const int* __restrict__ dst,
                                      const float* __restrict__ ew,
                                      const float* __restrict__ src_scale,
                                      float* __restrict__ agg) {
  constexpr int WIDTH = 4 << LOG2C;
  unsigned tid = blockIdx.x * blockDim.x + threadIdx.x;
  unsigned e = tid >> LOG2C;
  unsigned j = tid & ((1u << LOG2C) - 1u);
  if (e >= NE) return;
  int s = src[e];
  int d = dst[e];
  float c = ew[e];
  if (SCALE) c *= src_scale[s];
  const float4* hp = (const float4*)(h + (size_t)s * WIDTH);
  float4 v = hp[j];
  float* o = agg + (size_t)d * WIDTH + 4u * j;
  unsafeAtomicAdd(o + 0, v.x * c);
  unsafeAtomicAdd(o + 1, v.y * c);
  unsafeAtomicAdd(o + 2, v.z * c);
  unsafeAtomicAdd(o + 3, v.w * c);
}

// out[NN,FOUT] = act( (A * diag(row_scale)) @ W[KD,FOUT] + bias )
// One wave per 16x16 C tile; blockDim = 32 * (FOUT/16); grid = NN/16.
// A tile (16 x 128, pre-scaled) staged in LDS via float4, shared by all waves.
// FOUT is compile-time so every B offset (k0*FOUT) is an immediate off one base ptr.
// Fragment layouts per CDNA5 ISA 7.12.2:
//   A (16x4 f32): lane holds (M=lane&15, K=k0+2*(lane>>4)) and next K.
//   B (4x16 f32): VGPR v, lanes 0-15 -> row k0+v, lanes 16-31 -> row k0+v+2.
//   C/D (16x16):  VGPR v, lanes 0-15 -> M=v, lanes 16-31 -> M=v+8; N=lane&15.
template <int FOUT, bool RELU, bool HAS_BIAS>
__global__ void wmma_gemm_kernel(const float* __restrict__ A,
                                 const float* __restrict__ row_scale,
                                 const float* __restrict__ W,
                                 const float* __restrict__ bias,
                                 float* __restrict__ out) {
  __shared__ float As[16 * KD];  // 8 KB
  const int m_base = blockIdx.x * 16;

  // Cooperative staging: 512 float4 (global_load_b128 -> ds_store_b128), row-scaled.
  {
    constexpr int NVEC = 16 * KD / 4;  // 512
    for (int v = threadIdx.x; v < NVEC; v += blockDim.x) {
      int r = v >> 5;              // 32 float4 per row
      int cv = v & 31;
      float4 x = ((const float4*)(A + (size_t)(m_base + r) * KD))[cv];
      float s = row_scale[m_base + r];
      x.x *= s; x.y *= s; x.z *= s; x.w *= s;
      *(float4*)&As[r * KD + 4 * cv] = x;
    }
  }
  __syncthreads();

  const int wave = threadIdx.x >> 5;
  const int lane = threadIdx.x & 31;
  const int n_base = wave * 16;
  const int lo = lane & 15;
  const int hi = lane >> 4;

  v8f acc = {};
  // Per-lane B base: column (n_base+lo), starting row 2*hi; all k offsets constant.
  const float* bbase = W + (size_t)(2 * hi) * FOUT + n_base + lo;
  const float* abase = &As[lo * KD + 2 * hi];

  #pragma unroll
  for (int k0 = 0; k0 < KD; k0 += 4) {
    v2f a = *(const v2f*)(abase + k0);       // ds_load_b64 (8B aligned)
    v2f b;
    b.x = bbase[(size_t)(k0 + 0) * FOUT];    // immediate-offset global_load_b32
    b.y = bbase[(size_t)(k0 + 1) * FOUT];
    acc = __builtin_amdgcn_wmma_f32_16x16x4_f32(
        /*neg_a=*/false, a, /*neg_b=*/false, b,
        /*c_mod=*/(short)0, acc, /*reuse_a=*/false, /*reuse_b=*/false);
  }

  const int col = n_base + lo;
  const float bv = HAS_BIAS ? bias[col] : 0.0f;
  #pragma unroll
  for (int v = 0; v < 8; ++v) {
    int row = m_base + v + 8 * hi;
    float val = acc[v] + bv;
    if (RELU) val = val > 0.0f ? val : 0.0f;
    out[(size_t)row * FOUT + col] = val;
  }
}

// out[n,f] = agg[n,f] * in_norm[n] + b2[f]   (final layer epilogue, fout=64)
__global__ void final_epilogue_kernel(const float* __restrict__ agg,
                                      const float* __restrict__ in_norm,
                                      const float* __restrict__ b2,
                                      float* __restrict__ out) {
  int i = blockIdx.x * blockDim.x + threadIdx.x;
  if (i >= NN * 64) return;
  int n = i >> 6;
  int f = i & 63;
  out[i] = agg[i] * in_norm[n] + b2[f];
}

extern "C" void kernel_launch(void* const* d_in, const int* in_sizes, int n_in,
                              void* d_out, int out_size, void* d_ws, size_t ws_size,
                              hipStream_t stream) {
  const float* features = (const float*)d_in[0];
  const int*   src      = (const int*)d_in[1];
  const int*   dst      = (const int*)d_in[2];
  const float* ew       = (const float*)d_in[3];
  const float* W0 = (const float*)d_in[4];
  const float* b0 = (const float*)d_in[5];
  const float* W1 = (const float*)d_in[6];
  const float* b1 = (const float*)d_in[7];
  const float* W2 = (const float*)d_in[8];
  const float* b2 = (const float*)d_in[9];
  float* out = (float*)d_out;

  // Workspace layout (floats): out_norm[NN] | in_norm[NN] | bufA[NN*128] | bufB[NN*128]
  float* ws       = (float*)d_ws;
  float* out_norm = ws;
  float* in_norm  = ws + NN;
  float* bufA     = ws + 2 * NN;
  float* bufB     = bufA + (size_t)NN * KD;

  // Degrees -> norms (computed in place)
  fill_zero_kernel<<<512, 256, 0, stream>>>(out_norm, 2 * NN);
  degree_kernel<<<(NE + 255) / 256, 256, 0, stream>>>(src, dst, out_norm, in_norm);
  norm_kernel<<<(2 * NN + 255) / 256, 256, 0, stream>>>(out_norm, 2 * NN);

  // Layer 0: aggregate (width 128, out_norm fused into edge coeff), then GEMM+bias+relu
  fill_zero_kernel<<<2048, 256, 0, stream>>>(bufA, NN * KD);
  edge_aggregate_kernel<5, true><<<(NE * 32) / 256, 256, 0, stream>>>(
      features, src, dst, ew, out_norm, bufA);
  wmma_gemm_kernel<128, true, true><<<NN / 16, 256, 0, stream>>>(
      bufA, in_norm, W0, b0, bufB);

  // Layer 1: same shape
  fill_zero_kernel<<<2048, 256, 0, stream>>>(bufA, NN * KD);
  edge_aggregate_kernel<5, true><<<(NE * 32) / 256, 256, 0, stream>>>(
      bufB, src, dst, ew, out_norm, bufA);
  wmma_gemm_kernel<128, true, true><<<NN / 16, 256, 0, stream>>>(
      bufA, in_norm, W1, b1, bufB);

  // Layer 2 (128 -> 64): GEMM first (out_norm as row scale, no bias/relu),
  // then aggregate at width 64, then in_norm + bias epilogue.
  wmma_gemm_kernel<64, false, false><<<NN / 16, 128, 0, stream>>>(
      bufB, out_norm, W2, nullptr, bufA);
  fill_zero_kernel<<<2048, 256, 0, stream>>>(bufB, NN * 64);
  edge_aggregate_kernel<4, false><<<(NE * 16) / 256, 256, 0, stream>>>(
      bufA, src, dst, ew, nullptr, bufB);
  final_epilogue_kernel<<<(NN * 64 + 255) / 256, 256, 0, stream>>>(bufB, in_norm, b2, out);
}